// LTCStyleLayer_44710609551595
// MI455X (gfx1250) — compile-verified
//
#include <hip/hip_runtime.h>
#include <math.h>

typedef float v2f __attribute__((ext_vector_type(2)));
typedef float v8f __attribute__((ext_vector_type(8)));

#define B_ 64
#define T_ 512
#define D_ 128
#define H_ 1024

// One wave computes one 16x16 tile of h_t.
//   z   = seq[:,t,:] @ W_in^T + h_{t-1} @ W_rec^T + b_in + bias
//   h_t = h_{t-1} + (tanh(z) - h_{t-1}) * (DT / tau),  DT = 1
// h_{t-1} is read from out[:, t-1, :] (written by the previous launch);
// h_t is written to out[:, t, :].
__global__ __launch_bounds__(256)
void ltc_step_kernel(const float* __restrict__ seq,
                     const float* __restrict__ W_in,
                     const float* __restrict__ b_in,
                     const float* __restrict__ W_rec,
                     const float* __restrict__ bias,
                     const float* __restrict__ tau_p,
                     float* __restrict__ out,   // [B, T, H]
                     int t)
{
    const int wave = threadIdx.x >> 5;
    const int lane = threadIdx.x & 31;
    const int half = lane >> 4;      // 0: lanes 0-15, 1: lanes 16-31
    const int r16  = lane & 15;

    const int tile = blockIdx.x * 8 + wave;   // 256 tiles total
    const int n0 = (tile & 63) << 4;          // H (output feature) tile
    const int m0 = (tile >> 6) << 4;          // B (batch) tile

    const float inv_tau = 1.0f / tau_p[0];

    v8f acc = {};

    // ---- input projection: A[m,k] = seq[(m0+m)*T*D + t*D + k],
    //      B[k,n] = W_in[(n0+n)*D + k]  (B = W_in^T, columns contiguous)
    {
        const float* Arow = seq  + (size_t)(m0 + r16) * (T_ * D_) + (size_t)t * D_;
        const float* Brow = W_in + (size_t)(n0 + r16) * D_;
        #pragma unroll 8
        for (int k = 0; k < D_; k += 4) {
            v2f a = *(const v2f*)(Arow + k + 2 * half);  // lane<16: k,k+1 ; lane>=16: k+2,k+3
            v2f b = *(const v2f*)(Brow + k + 2 * half);
            acc = __builtin_amdgcn_wmma_f32_16x16x4_f32(
                false, a, false, b, (short)0, acc, false, false);
        }
    }

    // ---- recurrence: A[m,k] = out[(m0+m)*T*H + (t-1)*H + k],
    //      B[k,n] = W_rec[(n0+n)*H + k]
    if (t > 0) {
        const float* Arow = out   + (size_t)(m0 + r16) * (T_ * H_) + (size_t)(t - 1) * H_;
        const float* Brow = W_rec + (size_t)(n0 + r16) * H_;
        #pragma unroll 8
        for (int k = 0; k < H_; k += 4) {
            v2f a = *(const v2f*)(Arow + k + 2 * half);
            v2f b = *(const v2f*)(Brow + k + 2 * half);
            acc = __builtin_amdgcn_wmma_f32_16x16x4_f32(
                false, a, false, b, (short)0, acc, false, false);
        }
    }

    // ---- epilogue. C/D layout: VGPR r, lanes<16 -> M=r, lanes>=16 -> M=r+8; N=lane&15.
    const int col = n0 + r16;
    const float bsum = b_in[col] + bias[col];
    #pragma unroll
    for (int r = 0; r < 8; ++r) {
        const int row = m0 + r + 8 * half;
        const size_t base = (size_t)row * (T_ * H_);
        float hprev = (t > 0) ? out[base + (size_t)(t - 1) * H_ + col] : 0.0f;
        float z = acc[r] + bsum;
        float f = tanhf(z);
        out[base + (size_t)t * H_ + col] = hprev + (f - hprev) * inv_tau;
    }
}

// h_last[b, :] = out[b, T-1, :]
__global__ __launch_bounds__(256)
void ltc_hlast_kernel(const float* __restrict__ out, float* __restrict__ hlast)
{
    int i = blockIdx.x * 256 + threadIdx.x;       // 0 .. B*H/4 - 1  (16384)
    int b = i >> 8;                                // H/4 = 256 float4 per row
    int r = i & 255;
    const float4* src = (const float4*)(out + (size_t)b * (T_ * H_) + (size_t)(T_ - 1) * H_);
    float4* dst = (float4*)(hlast + (size_t)b * H_);
    dst[r] = src[r];
}

extern "C" void kernel_launch(void* const* d_in, const int* in_sizes, int n_in,
                              void* d_out, int out_size, void* d_ws, size_t ws_size,
                              hipStream_t stream) {
    (void)in_sizes; (void)n_in; (void)d_ws; (void)ws_size; (void)out_size;

    const float* seq   = (const float*)d_in[0];  // [B,T,D]
    const float* W_in  = (const float*)d_in[1];  // [H,D]
    const float* b_in  = (const float*)d_in[2];  // [H]
    const float* W_rec = (const float*)d_in[3];  // [H,H]
    const float* bias  = (const float*)d_in[4];  // [H]
    const float* tau   = (const float*)d_in[5];  // scalar

    float* out   = (float*)d_out;                      // [B,T,H]
    float* hlast = out + (size_t)B_ * T_ * H_;         // [B,H]

    // 512 dependent steps; kernel boundaries provide the device-wide sync on h.
    for (int t = 0; t < T_; ++t) {
        ltc_step_kernel<<<32, 256, 0, stream>>>(seq, W_in, b_in, W_rec, bias, tau, out, t);
    }
    ltc_hlast_kernel<<<64, 256, 0, stream>>>(out, hlast);
}